// MorphologicalDegradation_6408091206301
// MI455X (gfx1250) — compile-verified
//
#include <hip/hip_runtime.h>
#include <stdint.h>

// ---------------------------------------------------------------------------
// MorphologicalDegradation for MI455X (gfx1250).
// Memory-bound f32 product-stencil (circular structuring element, k in
// {3,5,7,9}). Roofline: ~70 MB HBM traffic => ~3 us at 23.3 TB/s; compute
// ~120-290 M f32 mults, same order => keep both lean.
// CDNA5 path: pad pass into d_ws, then per-workgroup TDM tensor_load_to_lds
// DMA of a 24x72 f32 tile into LDS (TENSORcnt-tracked), fully-unrolled
// VALU tap products out of LDS, b128 stores.
// ---------------------------------------------------------------------------

#define IMG_H 512
#define IMG_W 512
#define NIMG  16
#define PADR  4                      // max radius (k=9)
#define PH    (IMG_H + 2 * PADR)     // 520 padded rows
#define PW    528                    // 520 cols rounded up to 16 for alignment
#define TW    64                     // output tile width per workgroup
#define TH    16                     // output tile height per workgroup
#define LW    (TW + 2 * PADR)        // 72 LDS tile width
#define LH    (TH + 2 * PADR)        // 24 LDS tile height

typedef unsigned int       u32;
typedef unsigned long long u64;
typedef __attribute__((ext_vector_type(4))) u32 v4u;
typedef __attribute__((ext_vector_type(4))) int v4i;
typedef __attribute__((ext_vector_type(8))) int v8i;

// Pass 1: zero-padded copy of all 16 images into workspace (PH x PW each).
// Makes every TDM tile read physically in-bounds (TDM cannot clip negative
// tile offsets); zero halo also provides the reference's zero-pad semantics.
__global__ __launch_bounds__(256) void morph_pad_kernel(
    const float* __restrict__ x, float* __restrict__ xp, int total4) {
  int idx = blockIdx.x * blockDim.x + threadIdx.x;
  if (idx >= total4) return;
  int q   = idx % (PW / 4);
  int t   = idx / (PW / 4);
  int py  = t % PH;
  int img = t / PH;
  int sy  = py - PADR;
  float4 o = make_float4(0.f, 0.f, 0.f, 0.f);
  if (sy >= 0 && sy < IMG_H) {
    const float* src = &x[((size_t)img * IMG_H + sy) * IMG_W];
#pragma unroll
    for (int j = 0; j < 4; ++j) {
      int sx = 4 * q + j - PADR;
      ((float*)&o)[j] = (sx >= 0 && sx < IMG_W) ? src[sx] : 0.0f;
    }
  }
  *(float4*)&xp[(size_t)idx * 4] = o;   // 16B-aligned (PW % 4 == 0)
}

// Fully-unrolled circular-tap product for compile-time radius C.
// Each thread computes 4 horizontally adjacent pixels -> one b128 store;
// overlapping tap windows get CSE'd into shared (vector) LDS loads.
template <int C>
__device__ __forceinline__ void morph_compute(
    const float* __restrict__ tile, float* __restrict__ out,
    int img, int oy0, int ox0, float sgn, float basec, float weight) {
  const int lxq = (int)(threadIdx.x & 15);   // 16 quads of 4 pixels per row
  const int ly  = (int)(threadIdx.x >> 4);   // 16 rows
  const float* ctr = &tile[(ly + PADR) * LW + (4 * lxq + PADR)];

  float r[4];
#pragma unroll
  for (int p = 0; p < 4; ++p) {
    float acc = 1.0f;
#pragma unroll
    for (int dy = -C; dy <= C; ++dy) {
#pragma unroll
      for (int dx = -C; dx <= C; ++dx) {
        if (dx * dx + dy * dy <= C * C) {        // compile-time membership
          acc *= fmaf(sgn, ctr[dy * LW + dx + p], basec);
        }
      }
    }
    float morphed = fmaf(sgn, acc, basec);       // erode: acc ; dilate: 1-acc
    morphed = fminf(1.0f, fmaxf(0.0f, morphed));
    float xc = ctr[p];
    float v  = fmaf(weight, morphed - xc, xc);   // (1-w)*x + w*morphed
    r[p] = fminf(1.0f, fmaxf(0.0f, v));
  }
  float4 o = make_float4(r[0], r[1], r[2], r[3]);
  *(float4*)&out[((size_t)img * IMG_H + (oy0 + ly)) * IMG_W + (ox0 + 4 * lxq)] = o;
}

// Pass 2: one TDM DMA per workgroup loads the halo'd tile into LDS, then the
// radius-specialized unrolled compute runs out of LDS.
__global__ __launch_bounds__(256) void morph_main_kernel(
    const float* __restrict__ xp, const int* __restrict__ ip,
    float* __restrict__ out) {
  __shared__ float tile[LH * LW];   // 24 x 72 f32 = 6912 B

  const int ox0 = blockIdx.x * TW;
  const int oy0 = blockIdx.y * TH;
  const int img = blockIdx.z;

  // ---- TDM issue: scalar (SGPR) branch so only wave 0 reaches the
  // instruction (TDM ignores EXEC, so a merely EXEC-masked path would DMA 8x).
  if (__builtin_amdgcn_readfirstlane((int)threadIdx.x) == 0) {
    u64 gaddr = (u64)(uintptr_t)(xp + ((size_t)img * PH + oy0) * PW + ox0);
    u32 ldsa  = (u32)(uintptr_t)(&tile[0]);  // flat addr low 32 bits == LDS offset

    // D# group 0: count=1 | lds_addr | global_addr[56:0] | type=2
    v4u g0;
    g0.x = 1u;
    g0.y = ldsa;
    g0.z = (u32)gaddr;
    g0.w = ((u32)(gaddr >> 32) & 0x01FFFFFFu) | 0x80000000u;  // type=2 @ [127:126]

    // D# group 1: data_size=4B, 2D tile 72x24, tensor_dim0_stride=PW,
    // tensor dims set huge (1<<20) so OOB zero-fill never triggers
    // (padding pass guarantees physical in-bounds reads).
    u32 w0 = (2u << 16);                       // data_size=2 (4 bytes)
    u32 w1 = 0u;                               // barrier addr 0; TD0[15:0]=0
    u32 w2 = 16u;                              // TD0[31:16]=16 (TD0=1<<20); TD1 lo=0
    u32 w3 = 16u | ((u32)LW << 16);            // TD1[31:16]=16; tile_dim0=72
    u32 w4 = (u32)LH;                          // tile_dim1=24, tile_dim2=0
    u32 w5 = (u32)PW;                          // tensor_dim0_stride=528 elements
    u32 w6 = 0u;                               // stride0 hi, stride1 lo = 0
    u32 w7 = 0u;                               // stride1 hi = 0
    v8i g1;
    g1[0] = (int)w0; g1[1] = (int)w1; g1[2] = (int)w2; g1[3] = (int)w3;
    g1[4] = (int)w4; g1[5] = (int)w5; g1[6] = (int)w6; g1[7] = (int)w7;

    v4i z4 = (v4i)0;  // groups 2/3 zero => tile_dim2/3/4 = 0 (2D tensor)

#if defined(__clang_major__) && (__clang_major__ >= 23)
    v8i z8 = (v8i)0;
    __builtin_amdgcn_tensor_load_to_lds(g0, g1, z4, z4, z8, 0);
#else
    __builtin_amdgcn_tensor_load_to_lds(g0, g1, z4, z4, 0);
#endif
    __builtin_amdgcn_s_wait_tensorcnt(0);      // DMA data landed in LDS
  }
  __syncthreads();                             // publish tile to all 8 waves

  // ---- Morphology parameters computed on-device from input i (uniform).
  int i  = __builtin_amdgcn_readfirstlane(ip[0]);
  int ii = (i < 49) ? i : 49;
  float kf = 3.0f + (6.0f * (float)ii) / 49.0f;   // linspace(3,9,50)[ii]
  int k = (int)kf;
  if ((k & 1) == 0) k++;                          // odd-ify => k in {3,5,7,9}
  int c = k >> 1;                                 // radius 1..4
  float intensity = (float)(i + 1) / 50.0f;
  intensity = fminf(1.0f, fmaxf(0.0f, intensity));
  bool  erode  = intensity < 0.5f;
  float weight = erode ? intensity * 2.0f : (intensity - 0.5f) * 2.0f;
  float basec  = erode ? 0.0f : 1.0f;             // tap term = basec + sgn*v
  float sgn    = erode ? 1.0f : -1.0f;            // erode: v ; dilate: 1-v

  switch (c) {                                    // radius-specialized unroll
    case 1:  morph_compute<1>(tile, out, img, oy0, ox0, sgn, basec, weight); break;
    case 2:  morph_compute<2>(tile, out, img, oy0, ox0, sgn, basec, weight); break;
    case 3:  morph_compute<3>(tile, out, img, oy0, ox0, sgn, basec, weight); break;
    default: morph_compute<4>(tile, out, img, oy0, ox0, sgn, basec, weight); break;
  }
}

extern "C" void kernel_launch(void* const* d_in, const int* in_sizes, int n_in,
                              void* d_out, int out_size, void* d_ws, size_t ws_size,
                              hipStream_t stream) {
  (void)in_sizes; (void)n_in; (void)out_size; (void)ws_size;
  const float* x  = (const float*)d_in[0];
  const int*   ip = (const int*)d_in[1];
  float* out = (float*)d_out;
  float* xp  = (float*)d_ws;     // needs NIMG*PH*PW*4 = ~17.6 MB of scratch

  // Pass 1: build zero-padded copy (fully overwritten every call).
  int total4 = NIMG * PH * (PW / 4);
  int blocks = (total4 + 255) / 256;
  morph_pad_kernel<<<blocks, 256, 0, stream>>>(x, xp, total4);

  // Pass 2: TDM-tiled morphology.
  dim3 grid(IMG_W / TW, IMG_H / TH, NIMG);   // 8 x 32 x 16
  morph_main_kernel<<<grid, 256, 0, stream>>>(xp, ip, out);
}